// C4MoETop1_62380105007500
// MI455X (gfx1250) — compile-verified
//
#include <hip/hip_runtime.h>
#include <hip/hip_bf16.h>

// ---------------------------------------------------------------------------
// C4MoETop1 soft-VM step for MI455X (gfx1250, wave32).
//
// Roofline: the reference's read_memory() 512-wide soft-scan collapses to a
// 5-tap gather (eq_gate < 1e-16 beyond |d|=2 at SCALE=20, integer addrs), so
// the kernel moves only ~8 MB -> far below the 23.3 TB/s HBM roofline; it is
// VALU/latency bound. Therefore:
//   * gather windows staged via GLOBAL_LOAD_ASYNC_TO_LDS (ASYNCcnt),
//     overlapped with ax-only gate computation,
//   * div_expert = W[16x64] x q[64] mat-vec done with 16 chained
//     V_WMMA_F32_16X16X4_F32 per pass (f32 accumulate), 2 passes per wave,
//   * sigmoids use v_exp_f32 + v_rcp_f32 (1-ULP) instead of IEEE divide
//     sequences; all /2^k divides folded to exact multiplies.
// ---------------------------------------------------------------------------

typedef __attribute__((ext_vector_type(2))) float v2f;
typedef __attribute__((ext_vector_type(8))) float v8f;

#define BLOCK 256
#define MEMSZ 512

#if __has_builtin(__builtin_amdgcn_global_load_async_to_lds_b32) && \
    __has_builtin(__builtin_amdgcn_s_wait_asynccnt)
#define USE_ASYNC 1
typedef __attribute__((address_space(1))) int glb_int;
typedef __attribute__((address_space(3))) int lds_int;
#else
#define USE_ASYNC 0
#endif

#if __has_builtin(__builtin_amdgcn_wmma_f32_16x16x4_f32)
#define USE_WMMA 1
#else
#define USE_WMMA 0
#endif

__device__ __forceinline__ float fast_rcp(float x) {
#if __has_builtin(__builtin_amdgcn_rcpf)
  return __builtin_amdgcn_rcpf(x);  // v_rcp_f32, ~1 ULP
#else
  return 1.0f / x;
#endif
}
__device__ __forceinline__ float siluf(float x) {
  return x * fast_rcp(1.0f + __expf(-x));
}
// (silu(SCALE*x + 0.5*SCALE) - silu(SCALE*x - 0.5*SCALE)) / SCALE, SCALE=20
__device__ __forceinline__ float silu_thr(float x) {
  return (siluf(20.0f * x + 10.0f) - siluf(20.0f * x - 10.0f)) * 0.05f;
}
__device__ __forceinline__ float eq_gate(float a, float b) {
  float d = a - b;
  return silu_thr(d + 0.5f) * silu_thr(-d + 0.5f);
}
__device__ __forceinline__ float ge_gate(float a, float b) {
  return silu_thr(a - b + 0.5f);
}
__device__ __forceinline__ float gt_gate(float a, float b) {
  return silu_thr(a - b - 0.5f);
}
__device__ __forceinline__ float swiglu(float a, float b) {
  return a * siluf(b) - a * siluf(-b);
}
__device__ __forceinline__ float lane_bcast(float v, int src_lane) {
  int r = __builtin_amdgcn_ds_bpermute(src_lane << 2, __float_as_int(v));
  return __int_as_float(r);
}

__global__ __launch_bounds__(BLOCK) void C4MoETop1_62380105007500_kernel(
    const int* __restrict__ pc_i, const int* __restrict__ sp_i,
    const int* __restrict__ bp_i, const float* __restrict__ ax_i,
    const float* __restrict__ memory, float* __restrict__ out) {
  const int tid  = blockIdx.x * BLOCK + threadIdx.x;  // grid sized exactly
  const int lane = threadIdx.x & 31;
  const float* row = memory + (size_t)tid * MEMSZ;

  const int   pcI = pc_i[tid];
  const int   spI = sp_i[tid];
  const float pcF = (float)pcI;
  const float spF = (float)spI;
  const float bpF = (float)bp_i[tid];
  const float ax  = ax_i[tid];

  // ---- stage the two 5-tap gather windows (pc, sp) --------------------------
  int idxs[10];
#pragma unroll
  for (int d = 0; d < 5; ++d) {
    int a = pcI + d - 2, b = spI + d - 2;
    idxs[d]     = min(max(a, 0), MEMSZ - 1);
    idxs[5 + d] = min(max(b, 0), MEMSZ - 1);
  }
  float win[10];
#if USE_ASYNC
  __shared__ float stage[BLOCK * 10];
#pragma unroll
  for (int j = 0; j < 10; ++j) {
    __builtin_amdgcn_global_load_async_to_lds_b32(
        (glb_int*)(row + idxs[j]),
        (lds_int*)&stage[threadIdx.x * 10 + j], 0, 0);
  }
#else
#pragma unroll
  for (int j = 0; j < 10; ++j) win[j] = row[idxs[j]];
#endif

  // ---- ax-only work overlaps the async gather -------------------------------
  const float ax_safe = ax + eq_gate(ax, 0.0f) * 0.001f;

  float sg[32];  // shift gates, depend only on ax
#pragma unroll
  for (int c = 0; c < 32; ++c) {
    float d = ax - (float)c;
    sg[c] = swiglu(silu_thr(d + 0.5f), silu_thr(-d + 0.5f));
  }

#if USE_ASYNC
  __builtin_amdgcn_s_wait_asynccnt(0);
  asm volatile("" ::: "memory");
#pragma unroll
  for (int j = 0; j < 10; ++j) win[j] = stage[threadIdx.x * 10 + j];
#endif

  // ---- soft memory reads (exact within fp32 of the 512-wide reference scan)
  float instr = 0.0f, st = 0.0f;
#pragma unroll
  for (int d = 0; d < 5; ++d) {
    int ia = pcI + d - 2, ib = spI + d - 2;
    float ga = (ia >= 0 && ia < MEMSZ) ? eq_gate(pcF, (float)ia) : 0.0f;
    float gb = (ib >= 0 && ib < MEMSZ) ? eq_gate(spF, (float)ib) : 0.0f;
    instr += ga * win[d];
    st    += gb * win[5 + d];
  }

  const float opcode = instr - 256.0f * floorf(instr * (1.0f / 256.0f));
  const float imm    = floorf(instr * (1.0f / 256.0f));

  // ---- expert scores + argmax (first max, like jnp.argmax) ------------------
  float best = -1.0f;
  int   sel  = 0;
#pragma unroll
  for (int e = 0; e < 40; ++e) {
    float s = eq_gate(opcode, (float)e);
    if (s > best) { best = s; sel = e; }
  }

  // ---- div_expert: divr_i = sum_q (th1-th2)[i,q] * q  via WMMA f32 16x16x4 --
  float divr;
#if USE_WMMA
  {
    const int h = (lane >> 4) & 1;  // K-half owned by this lane (A layout)
    float dvp[2];
#pragma unroll
    for (int p = 0; p < 2; ++p) {
      const int src = p * 16 + (lane & 15);
      const float stp  = lane_bcast(st, src);
      const float axp  = lane_bcast(ax_safe, src);
      v8f acc = {0.f, 0.f, 0.f, 0.f, 0.f, 0.f, 0.f, 0.f};
#pragma unroll
      for (int c4 = 0; c4 < 16; ++c4) {
        const int   k0 = c4 * 4 + h * 2;
        const float q0 = (float)k0, q1 = (float)(k0 + 1);
        float t0 = stp - q0 * axp + 0.5f;
        float t1 = stp - q1 * axp + 0.5f;
        float w0 = silu_thr(t0) - silu_thr(t0 - axp);  // th1 - th2 at q0
        float w1 = silu_thr(t1) - silu_thr(t1 - axp);  // th1 - th2 at q1
        v2f a = {w0, w1};
        v2f b = {q0, q1};  // B[k,j] = q_k for every column j
        acc = __builtin_amdgcn_wmma_f32_16x16x4_f32(
            false, a, false, b, (short)0, acc, false, false);
      }
      // D row i is divr_i broadcast across columns; pick diagonal + bpermute.
      const int r = lane & 7;
      float diag = acc[0];
      diag = (r == 1) ? acc[1] : diag;
      diag = (r == 2) ? acc[2] : diag;
      diag = (r == 3) ? acc[3] : diag;
      diag = (r == 4) ? acc[4] : diag;
      diag = (r == 5) ? acc[5] : diag;
      diag = (r == 6) ? acc[6] : diag;
      diag = (r == 7) ? acc[7] : diag;
      const int e  = lane & 15;
      const int sl = (e < 8) ? e : (e + 8);  // lanes 16..23 hold rows 8..15
      dvp[p] = lane_bcast(diag, sl);
    }
    divr = (lane < 16) ? dvp[0] : dvp[1];
  }
#else
  {
    divr = 0.0f;
#pragma unroll
    for (int q = 0; q < 64; ++q) {
      float t1 = st - (float)q * ax_safe + 0.5f;
      divr += (silu_thr(t1) - silu_thr(t1 - ax_safe)) * (float)q;
    }
  }
#endif

  const float modr = st - swiglu(divr, ax_safe);

  // ---- shifts (1/2^c exact; constant-folded after unroll) -------------------
  float shl = 0.0f, shr = 0.0f;
#pragma unroll
  for (int c = 0; c < 32; ++c) {
    float pw   = (float)(1u << c);
    float ipw  = 1.0f / (float)(1u << c);  // exact power of two
    shl += swiglu(st, pw) * sg[c];
    shr += floorf(st * ipw) * sg[c];
  }

  // ---- bitwise --------------------------------------------------------------
  float and_o = 0.0f, or_o = 0.0f, xor_o = 0.0f;
#pragma unroll
  for (int k = 0; k < 16; ++k) {
    float pb  = (float)(1u << k);
    float ipb = 1.0f / (float)(1u << k);  // exact power of two
    float da = floorf(st * ipb);
    float ba = da - 2.0f * floorf(da * 0.5f);
    float db = floorf(ax * ipb);
    float bb = db - 2.0f * floorf(db * 0.5f);
    float ab = swiglu(ba, bb);
    and_o += ab * pb;
    or_o  += (ba + bb - ab) * pb;
    xor_o += (ba + bb - 2.0f * ab) * pb;
  }

  const float eqo = eq_gate(st, ax);

  // ---- select expert output (cndmask chain, no scratch) ---------------------
  float outv = ax;  // experts 18..39 are noop(ax)
  outv = (sel == 0)  ? imm                 : outv;
  outv = (sel == 1)  ? (bpF + imm)         : outv;
  outv = (sel == 2)  ? (st + ax)           : outv;
  outv = (sel == 3)  ? (st - ax)           : outv;
  outv = (sel == 4)  ? swiglu(st, ax)      : outv;
  outv = (sel == 5)  ? divr                : outv;
  outv = (sel == 6)  ? modr                : outv;
  outv = (sel == 7)  ? shl                 : outv;
  outv = (sel == 8)  ? shr                 : outv;
  outv = (sel == 9)  ? and_o               : outv;
  outv = (sel == 10) ? or_o                : outv;
  outv = (sel == 11) ? xor_o               : outv;
  outv = (sel == 12) ? eqo                 : outv;
  outv = (sel == 13) ? (1.0f - eqo)        : outv;
  outv = (sel == 14) ? gt_gate(ax, st)     : outv;
  outv = (sel == 15) ? gt_gate(st, ax)     : outv;
  outv = (sel == 16) ? ge_gate(ax, st)     : outv;
  outv = (sel == 17) ? ge_gate(st, ax)     : outv;

  out[tid] = outv * best;
}

extern "C" void kernel_launch(void* const* d_in, const int* in_sizes, int n_in,
                              void* d_out, int out_size, void* d_ws,
                              size_t ws_size, hipStream_t stream) {
  (void)n_in; (void)d_ws; (void)ws_size; (void)out_size;
  const int*   pc  = (const int*)d_in[0];
  const int*   sp  = (const int*)d_in[1];
  const int*   bp  = (const int*)d_in[2];
  const float* ax  = (const float*)d_in[3];
  const float* mem = (const float*)d_in[4];
  float*       out = (float*)d_out;
  const int n = in_sizes[0];          // B = 131072, multiple of BLOCK
  const int blocks = n / BLOCK;       // exact cover: EXEC all-ones for WMMA
  C4MoETop1_62380105007500_kernel<<<blocks, BLOCK, 0, stream>>>(pc, sp, bp, ax,
                                                                mem, out);
}